// SphereConv3D_47957604827620
// MI455X (gfx1250) — compile-verified
//
#include <hip/hip_runtime.h>

// ---------------- problem constants (from reference setup_inputs) ----------
#define CIN   32
#define COUT  64
#define T_    16
#define H_    64
#define W_    128
#define K_    (CIN * 27)        // 864  GEMM reduction dim
#define M_    (T_ * H_ * W_)    // 131072 output voxels
#define ASTR  868               // LDS row stride (dwords): mult of 4 (16B align),
                                // 868 % 64 = 36 -> conflict-free b64 frag reads

typedef __attribute__((ext_vector_type(2))) float v2f;
typedef __attribute__((ext_vector_type(4))) float v4f;
typedef __attribute__((ext_vector_type(8))) float v8f;

// ---------------------------------------------------------------------------
// Prep: xT[(z,y,x),ci] = x[ci,(z,y,x)]   (corner gathers become contiguous)
//       wT[co, tap*32+ci] = weight[co,ci,tap]  (B fragments contiguous in k)
// ---------------------------------------------------------------------------
__global__ __launch_bounds__(256)
void sphere_prep(const float* __restrict__ x, const float* __restrict__ w,
                 float* __restrict__ xT, float* __restrict__ wT)
{
    int idx = blockIdx.x * 256 + threadIdx.x;
    if (idx < M_ * CIN) {                 // coalesced read of x
        int ci = idx / M_;
        int s  = idx - ci * M_;
        xT[s * CIN + ci] = x[idx];
    }
    if (idx < COUT * K_) {
        int co  = idx / K_;
        int r   = idx - co * K_;          // r = tap*32 + ci (dst k)
        int tap = r >> 5;
        int ci  = r & 31;
        wT[idx] = w[(co * CIN + ci) * 27 + tap];
    }
}

// ---------------------------------------------------------------------------
// Main: per block = 16 output voxels (one M-tile), 4 waves = 4 Cout tiles.
// Phase 1: trilinear-sample A panel (16 x 864) into LDS.
// Phase 2: 216x V_WMMA_F32_16X16X4_F32 per wave, bias, coalesced store.
// ---------------------------------------------------------------------------
__global__ __launch_bounds__(128)
void sphere_conv_wmma(const float* __restrict__ xT,
                      const float* __restrict__ wT,
                      const float* __restrict__ bias,
                      const float* __restrict__ grid,
                      float* __restrict__ out)
{
    __shared__ float Ash[16 * ASTR];        // 55,552 B  A panel
    __shared__ float Csh[4 * 16 * 16];      //  4,096 B  C staging (per wave)

    const int tid  = threadIdx.x;
    const int lane = tid & 31;
    const int wave = tid >> 5;
    const int m0   = blockIdx.x * 16;       // 16 consecutive voxels (same h,t row)

    // ---------------- Phase 1: sampling -> LDS ----------------
    for (int p = tid; p < 16 * 27; p += 128) {
        const int ml  = p / 27;
        const int tap = p - ml * 27;
        const int m   = m0 + ml;
        const int xw  = m & (W_ - 1);
        const int hy  = (m >> 7) & (H_ - 1);
        const int tz  = m >> 13;
        const int kt  = tap / 9, kh = (tap / 3) % 3, kw = tap % 3;

        const int gidx = (((tz * 3 + kt) * (H_ * 3) + (hy * 3 + kh)) * (W_ * 3)
                          + (xw * 3 + kw)) * 3;
        const float gx = grid[gidx + 0];
        const float gy = grid[gidx + 1];
        const float gz = grid[gidx + 2];

        // align_corners=True mapping
        const float ix = (gx + 1.f) * 0.5f * (float)(W_ - 1);
        const float iy = (gy + 1.f) * 0.5f * (float)(H_ - 1);
        const float iz = (gz + 1.f) * 0.5f * (float)(T_ - 1);
        const float fx = floorf(ix), fy = floorf(iy), fz = floorf(iz);
        const int   x0 = (int)fx,   y0 = (int)fy,   z0 = (int)fz;
        const float tx = ix - fx,   ty = iy - fy,   tf = iz - fz;

        const float* base[8];
        float wgt[8];
        #pragma unroll
        for (int c = 0; c < 8; ++c) {
            const int dx = c & 1, dy = (c >> 1) & 1, dz = c >> 2;
            const int cx = x0 + dx, cy = y0 + dy, cz = z0 + dz;
            float wc = (dx ? tx : 1.f - tx) * (dy ? ty : 1.f - ty)
                     * (dz ? tf : 1.f - tf);
            const bool valid = (cx >= 0) & (cx <= W_ - 1) & (cy >= 0)
                             & (cy <= H_ - 1) & (cz >= 0) & (cz <= T_ - 1);
            wgt[c] = valid ? wc : 0.f;      // zeros padding
            const int cxc = min(max(cx, 0), W_ - 1);
            const int cyc = min(max(cy, 0), H_ - 1);
            const int czc = min(max(cz, 0), T_ - 1);
            base[c] = xT + ((czc * H_ + cyc) * W_ + cxc) * CIN;
        }

        float* dst = &Ash[ml * ASTR + tap * CIN];     // 16B aligned
        #pragma unroll
        for (int ch = 0; ch < CIN; ch += 4) {
            v4f acc = {0.f, 0.f, 0.f, 0.f};
            #pragma unroll
            for (int c = 0; c < 8; ++c) {
                v4f v = *(const v4f*)(base[c] + ch);  // contiguous 16B (L2-hot)
                acc += wgt[c] * v;
            }
            *(v4f*)(dst + ch) = acc;
        }
    }
    __syncthreads();

    // ---------------- Phase 2: WMMA f32 16x16x4 ----------------
    // A frag: lane<16 -> row=lane, K{0,1}; lane>=16 -> row=lane-16, K{2,3}
    // B frag: lane<16 -> col=lane, K{0,1}; lane>=16 -> col=lane-16, K{2,3}
    const int col    = lane & 15;
    const int khalf  = (lane >> 4) << 1;       // 0 or 2
    const int nbase  = wave * 16;
    const float* ap  = &Ash[col * ASTR + khalf];
    const float* bp  = wT + (nbase + col) * K_ + khalf;

    v8f c = {};
    #pragma unroll 4
    for (int k0 = 0; k0 < K_; k0 += 4) {
        v2f a = *(const v2f*)(ap + k0);        // ds_load_b64, conflict-free
        v2f b = *(const v2f*)(bp + k0);        // global b64, L2-hot (216 KB)
        c = __builtin_amdgcn_wmma_f32_16x16x4_f32(
                /*neg_a=*/false, a, /*neg_b=*/false, b,
                /*c_mod=*/(short)0, c, /*reuse_a=*/false, /*reuse_b=*/false);
    }

    // ---------------- epilogue: bias + transpose-through-LDS store --------
    const float bv = bias[nbase + col];
    float* cs = &Csh[wave * 256];
    const int rbase = (lane >> 4) * 8;         // C/D layout: VGPR r, M=r+8*(lane/16)
    #pragma unroll
    for (int r = 0; r < 8; ++r)
        cs[col * 16 + rbase + r] = c[r] + bv;  // cs[n][m]
    __syncthreads();

    const int ocol = lane >> 1;                // Cout within tile
    const int half = lane & 1;                 // which 8-voxel half
    const float* src = cs + ocol * 16 + half * 8;
    v4f o0 = *(const v4f*)(src);
    v4f o1 = *(const v4f*)(src + 4);
    float* op = out + (nbase + ocol) * M_ + m0 + half * 8;
    *(v4f*)(op)     = o0;                      // coalesced 64B runs per Cout
    *(v4f*)(op + 4) = o1;
}

// ---------------------------------------------------------------------------
extern "C" void kernel_launch(void* const* d_in, const int* in_sizes, int n_in,
                              void* d_out, int out_size, void* d_ws, size_t ws_size,
                              hipStream_t stream)
{
    const float* x      = (const float*)d_in[0];   // (1,32,16,64,128)
    const float* weight = (const float*)d_in[1];   // (64,32,3,3,3)
    const float* bias   = (const float*)d_in[2];   // (64,)
    const float* grid   = (const float*)d_in[3];   // (1,48,192,384,3)
    float*       out    = (float*)d_out;           // (1,64,16,64,128)

    // workspace: xT (16.78 MB) + wT (0.22 MB)  ~= 17 MB
    float* xT = (float*)d_ws;
    float* wT = xT + (size_t)M_ * CIN;

    const int prepN = M_ * CIN;                    // 4,194,304 (covers COUT*K_ too)
    sphere_prep<<<(prepN + 255) / 256, 256, 0, stream>>>(x, weight, xT, wT);
    sphere_conv_wmma<<<M_ / 16, 128, 0, stream>>>(xT, wT, bias, grid, out);
}